// BahdanauAttentionLayer_40742059770267
// MI455X (gfx1250) — compile-verified
//
#include <hip/hip_runtime.h>
#include <hip/hip_bf16.h>

// Problem constants (from the reference)
#define BB   4
#define TQ   512
#define TK   512
#define DIN  256
#define DM   256

typedef __attribute__((ext_vector_type(2))) float v2f;
typedef __attribute__((ext_vector_type(8))) float v8f;

// CDNA5 hardware tanh (TRANS32 op). Raw asm bypasses compiler hazard
// handling, so include one independent op (v_nop) after the trans op
// per the ISA "1 independent op before result use" rule.
__device__ __forceinline__ float tanh_hw(float x) {
  float r;
  asm volatile("v_tanh_f32 %0, %1\n\tv_nop" : "=v"(r) : "v"(x));
  return r;
}

// ---------------------------------------------------------------------------
// Generic batched fp32 WMMA GEMM: C[bz] = A[bz] * B[bz]
// One wave (32 lanes) computes a 16x16 tile of C via V_WMMA_F32_16X16X4_F32,
// stepping K by 4 per instruction. Layouts per ISA 7.12.2:
//   A 16x4:  lanes 0-15 -> M=lane, VGPR0/1 = K0/K1 ; lanes 16-31 -> K2/K3
//   B 4x16:  VGPR0 = row k0+2*half, VGPR1 = row k0+2*half+1, N = lane&15
//   C 16x16: VGPR i -> row m0 + i + 8*half, col n0 + (lane&15)
// ---------------------------------------------------------------------------
__global__ __launch_bounds__(32) void wmma_gemm_f32_kernel(
    const float* __restrict__ A, const float* __restrict__ B,
    float* __restrict__ C, int K, int lda, int ldb, int ldc,
    unsigned long long sA, unsigned long long sB, unsigned long long sC) {
  const int lane = threadIdx.x;
  const int half = lane >> 4;   // 0 or 1
  const int l16  = lane & 15;
  const int n0 = blockIdx.x * 16;
  const int m0 = blockIdx.y * 16;
  const int bz = blockIdx.z;

  A += (size_t)bz * sA;
  B += (size_t)bz * sB;
  C += (size_t)bz * sC;

  const float* aRow = A + (size_t)(m0 + l16) * lda;

  v8f acc = (v8f){0.f, 0.f, 0.f, 0.f, 0.f, 0.f, 0.f, 0.f};

  for (int k0 = 0; k0 < K; k0 += 4) {
    const int ka = k0 + 2 * half;
    v2f a, b;
    a.x = aRow[ka];
    a.y = aRow[ka + 1];
    b.x = B[(size_t)ka * ldb + n0 + l16];
    b.y = B[(size_t)(ka + 1) * ldb + n0 + l16];
    acc = __builtin_amdgcn_wmma_f32_16x16x4_f32(
        /*neg_a=*/false, a, /*neg_b=*/false, b,
        /*c_mod=*/(short)0, acc, /*reuse_a=*/false, /*reuse_b=*/false);
  }

  float* cRow = C + (size_t)(m0 + 8 * half) * ldc + n0 + l16;
#pragma unroll
  for (int i = 0; i < 8; ++i) cRow[(size_t)i * ldc] = acc[i];
}

// ---------------------------------------------------------------------------
// Additive (Bahdanau) scores: attn_raw[b,q,k] = sum_d scale[d]*tanh(qp+kp)
// Block = 256 threads, tile = 16 q-rows x 32 k-rows, full D=256 in LDS.
// Rows padded to 257 floats so k-row accesses hit distinct LDS banks.
// Each thread owns 1 q-row x 2 k-cols -> 512 hw-tanh per thread.
// ---------------------------------------------------------------------------
__global__ __launch_bounds__(256) void scores_kernel(
    const float* __restrict__ qp, const float* __restrict__ kp,
    const float* __restrict__ scale, float* __restrict__ attn) {
  __shared__ float qs[16][257];
  __shared__ float ks[32][257];
  __shared__ float sc[DM];

  const int tid = threadIdx.x;
  const int b   = blockIdx.z;
  const int q0  = blockIdx.y * 16;
  const int k0  = blockIdx.x * 32;

  for (int i = tid; i < 16 * DM; i += 256) {
    const int r = i >> 8, c = i & 255;
    qs[r][c] = qp[((size_t)(b * TQ + q0 + r)) * DM + c];
  }
  for (int i = tid; i < 32 * DM; i += 256) {
    const int r = i >> 8, c = i & 255;
    ks[r][c] = kp[((size_t)(b * TK + k0 + r)) * DM + c];
  }
  sc[tid] = scale[tid];
  __syncthreads();

  const int qi = tid >> 4;          // 0..15
  const int kb = (tid & 15) * 2;    // 0,2,..,30

  float s0 = 0.f, s1 = 0.f;
  for (int d = 0; d < DM; ++d) {
    const float qv = qs[qi][d];
    const float w  = sc[d];
    s0 += w * tanh_hw(qv + ks[kb + 0][d]);
    s1 += w * tanh_hw(qv + ks[kb + 1][d]);
  }

  float* row = attn + ((size_t)b * TQ + q0 + qi) * TK + k0 + kb;
  row[0] = s0;
  row[1] = s1;
}

// ---------------------------------------------------------------------------
// In-place softmax over rows of length TK=512 (one block per (b,q) row).
// ---------------------------------------------------------------------------
__global__ __launch_bounds__(256) void softmax_kernel(float* __restrict__ attn) {
  __shared__ float red[256];
  const int tid = threadIdx.x;
  float* p = attn + (size_t)blockIdx.x * TK;

  float x0 = p[tid];
  float x1 = p[tid + 256];

  red[tid] = fmaxf(x0, x1);
  __syncthreads();
  for (int s = 128; s > 0; s >>= 1) {
    if (tid < s) red[tid] = fmaxf(red[tid], red[tid + s]);
    __syncthreads();
  }
  const float m = red[0];
  __syncthreads();

  const float L2E = 1.4426950408889634f;
  const float e0 = __builtin_amdgcn_exp2f((x0 - m) * L2E);
  const float e1 = __builtin_amdgcn_exp2f((x1 - m) * L2E);

  red[tid] = e0 + e1;
  __syncthreads();
  for (int s = 128; s > 0; s >>= 1) {
    if (tid < s) red[tid] += red[tid + s];
    __syncthreads();
  }
  const float inv = 1.0f / red[0];

  p[tid]       = e0 * inv;
  p[tid + 256] = e1 * inv;
}

// ---------------------------------------------------------------------------
// Host-side launch
// ---------------------------------------------------------------------------
extern "C" void kernel_launch(void* const* d_in, const int* in_sizes, int n_in,
                              void* d_out, int out_size, void* d_ws, size_t ws_size,
                              hipStream_t stream) {
  (void)in_sizes; (void)n_in; (void)out_size; (void)ws_size;

  const float* query = (const float*)d_in[0];  // [B,Tq,DIN]
  const float* value = (const float*)d_in[1];  // [B,Tk,DIN]
  const float* Wq    = (const float*)d_in[2];  // [DIN,DM]
  const float* Wv    = (const float*)d_in[3];  // [DIN,DM]
  const float* scale = (const float*)d_in[4];  // [DM]

  float* out  = (float*)d_out;                       // [B,Tq,DIN]
  float* attn = out + (size_t)BB * TQ * DIN;         // [B,Tq,Tk]

  float* qp = (float*)d_ws;                          // [B*Tq, DM]
  float* kp = qp + (size_t)BB * TQ * DM;             // [B*Tk, DM]

  // K1a: qp = query @ Wq   (M = B*Tq = 2048, N = DM = 256, K = DIN = 256)
  {
    dim3 grid(DM / 16, (BB * TQ) / 16, 1);
    wmma_gemm_f32_kernel<<<grid, 32, 0, stream>>>(
        query, Wq, qp, DIN, DIN, DM, DM, 0ULL, 0ULL, 0ULL);
  }
  // K1b: kp = value @ Wv
  {
    dim3 grid(DM / 16, (BB * TK) / 16, 1);
    wmma_gemm_f32_kernel<<<grid, 32, 0, stream>>>(
        value, Wv, kp, DIN, DIN, DM, DM, 0ULL, 0ULL, 0ULL);
  }
  // K2: raw additive scores -> attn region of d_out
  {
    dim3 grid(TK / 32, TQ / 16, BB);
    scores_kernel<<<grid, 256, 0, stream>>>(qp, kp, scale, attn);
  }
  // K3: softmax in place over attn rows
  {
    softmax_kernel<<<BB * TQ, 256, 0, stream>>>(attn);
  }
  // K4: out[b] = attn[b] @ value[b]   (M=Tq, N=DIN, K=Tk, batched over B)
  {
    dim3 grid(DIN / 16, TQ / 16, BB);
    wmma_gemm_f32_kernel<<<grid, 32, 0, stream>>>(
        attn, value, out, TK, TK, DIN, DIN,
        (unsigned long long)TQ * TK, (unsigned long long)TK * DIN,
        (unsigned long long)TQ * DIN);
  }
}